// ExtractRelevantPatchesLayer_68521908240709
// MI455X (gfx1250) — compile-verified
//
#include <hip/hip_runtime.h>
#include <hip/hip_bf16.h>

typedef __attribute__((ext_vector_type(2))) float v2f;
typedef __attribute__((ext_vector_type(4))) float v4f;
typedef __attribute__((ext_vector_type(8))) float v8f;

#define PATCH   64
#define NPATCH  256       // candidate patches per image
#define TOPK    32
#define IMG     1024
#define NBATCH  16

// ---------------------------------------------------------------------------
// Kernel A: 64x64 average pooling of the heatmap via V_WMMA_F32_16X16X4_F32.
// One wave32 per patch. Each iteration: one NT b128 load (2 rows' worth of
// data per wave, 512 B coalesced) feeding two WMMAs with an all-ones B.
// Because B is all ones, D[m][n] += sum_k A[m][k] and we reduce over ALL m
// at the end, so the mapping of loaded elements onto A's (M,K) slots is
// irrelevant — each lane feeds its own .xy / .zw directly, no lane swaps.
// ---------------------------------------------------------------------------
__global__ __launch_bounds__(256) void pool_wmma_kernel(
    const float* __restrict__ heatmap, float* __restrict__ avg) {
  const int wave = (int)((blockIdx.x * blockDim.x + threadIdx.x) >> 5);
  const int lane = (int)(threadIdx.x & 31);
  const int b    = wave >> 8;          // image
  const int n    = wave & 255;         // patch index within image
  const int gh   = n >> 4, gw = n & 15;

  const float* base = heatmap + ((size_t)b * IMG + (size_t)gh * PATCH) * IMG
                              + (size_t)gw * PATCH;

  const int m    = lane & 15;          // column group within a 64-float row
  const int rsel = lane >> 4;          // lanes 0-15: row r, lanes 16-31: row r+1

  v8f acc = {};
  v2f ones; ones.x = 1.0f; ones.y = 1.0f;

  for (int r = 0; r < PATCH; r += 2) {
    const float* p = base + (size_t)(r + rsel) * IMG + 4 * m;
    v4f v = __builtin_nontemporal_load((const v4f*)p);
    v2f a1; a1.x = v.x; a1.y = v.y;
    v2f a2; a2.x = v.z; a2.y = v.w;
    acc = __builtin_amdgcn_wmma_f32_16x16x4_f32(false, a1, false, ones,
                                                (short)0, acc, false, false);
    acc = __builtin_amdgcn_wmma_f32_16x16x4_f32(false, a2, false, ones,
                                                (short)0, acc, false, false);
  }

  // Every column of D is identical (B = ones). Lanes 0-15 hold M=0..7 in
  // acc[0..7]; lanes 16-31 hold M=8..15. Sum VGPRs, then fold across halves.
  float part = acc[0] + acc[1] + acc[2] + acc[3]
             + acc[4] + acc[5] + acc[6] + acc[7];
  float tot = part + __shfl_xor(part, 16);
  if (lane == 0) avg[wave] = tot * (1.0f / 4096.0f);
}

// ---------------------------------------------------------------------------
// Kernel B: per-image stable top-32 by rank counting (matches lax.top_k:
// descending values, ties broken by lower index). One block per image.
// ---------------------------------------------------------------------------
__global__ __launch_bounds__(NPATCH) void topk_kernel(
    const float* __restrict__ avg, int* __restrict__ topidx) {
  __shared__ float v[NPATCH];
  const int b = (int)blockIdx.x;
  const int i = (int)threadIdx.x;
  v[i] = avg[b * NPATCH + i];
  __syncthreads();
  const float vi = v[i];
  int rank = 0;
  for (int j = 0; j < NPATCH; ++j) {
    float vj = v[j];
    rank += (vj > vi) || (vj == vi && j < i);
  }
  if (rank < TOPK) topidx[b * TOPK + rank] = i;
}

// ---------------------------------------------------------------------------
// Kernel C: gather selected 64x64x3 patches. One block per output patch.
// Row = 768 contiguous bytes (48 x b128), 64 rows, src row pitch 12288 B.
// Non-temporal b128 load/store: data is touched exactly once.
// ---------------------------------------------------------------------------
__global__ __launch_bounds__(256) void gather_kernel(
    const float* __restrict__ image, const int* __restrict__ topidx,
    float* __restrict__ out) {
  const int o  = (int)blockIdx.x;       // 0..511, == b*32 + k
  const int b  = o >> 5;
  const int n  = topidx[o];             // selected patch index (uniform -> s_load)
  const int gh = n >> 4, gw = n & 15;

  const v4f* src = (const v4f*)(image +
      (((size_t)b * IMG + (size_t)gh * PATCH) * IMG + (size_t)gw * PATCH) * 3);
  v4f* dst = (v4f*)(out + (size_t)o * (PATCH * PATCH * 3));

  // 64 rows * 48 float4 = 3072 float4 per patch; 12 per thread.
  for (int f = (int)threadIdx.x; f < PATCH * 48; f += 256) {
    int row = f / 48;
    int col = f - row * 48;
    v4f val = __builtin_nontemporal_load(src + (size_t)row * (IMG * 3 / 4) + col);
    __builtin_nontemporal_store(val, dst + f);
  }
}

// ---------------------------------------------------------------------------
extern "C" void kernel_launch(void* const* d_in, const int* in_sizes, int n_in,
                              void* d_out, int out_size, void* d_ws, size_t ws_size,
                              hipStream_t stream) {
  const float* heatmap = (const float*)d_in[0];   // (16,1024,1024,1) f32
  const float* image   = (const float*)d_in[1];   // (16,1024,1024,3) f32
  float* out = (float*)d_out;                     // (512,64,64,3) f32

  float* avg   = (float*)d_ws;                                  // 4096 f32
  int*   topix = (int*)((char*)d_ws + NBATCH * NPATCH * sizeof(float)); // 512 int

  // 16*256 = 4096 patches, 1 wave each, 8 waves per 256-thread block.
  pool_wmma_kernel<<<(NBATCH * NPATCH) / 8, 256, 0, stream>>>(heatmap, avg);
  topk_kernel<<<NBATCH, NPATCH, 0, stream>>>(avg, topix);
  gather_kernel<<<NBATCH * TOPK, 256, 0, stream>>>(image, topix, out);
}